// ImprovedGenderAwareLoss_15333033246944
// MI455X (gfx1250) — compile-verified
//
#include <hip/hip_runtime.h>
#include <hip/hip_bf16.h>
#include <math.h>

typedef __attribute__((ext_vector_type(2))) float v2f;
typedef __attribute__((ext_vector_type(8))) float v8f;

#define BN 4096
#define DD 512
#define NCLS 7
#define LDS_STRIDE 516   // 512 + 4 pad floats -> conflict-free ds_load_b64
#define INV_T (1.0f / 0.07f)

// ---------------------------------------------------------------------------
// Kernel 1: row-normalize features.  grid = 4096 blocks x 256 threads.
// ---------------------------------------------------------------------------
__global__ __launch_bounds__(256) void norm_kernel(const float* __restrict__ f,
                                                   float* __restrict__ fn) {
    __shared__ float red[8];
    __shared__ float rinv_s;
    const int row = blockIdx.x;
    const int tid = threadIdx.x;

    const float2 v = ((const float2*)(f + (size_t)row * DD))[tid];
    float ss = v.x * v.x + v.y * v.y;
    #pragma unroll
    for (int off = 1; off < 32; off <<= 1) ss += __shfl_xor(ss, off, 32);
    if ((tid & 31) == 0) red[tid >> 5] = ss;
    __syncthreads();
    if (tid < 8) {
        float t = red[tid];
        #pragma unroll
        for (int off = 1; off < 8; off <<= 1) t += __shfl_xor(t, off, 32);
        if (tid == 0) rinv_s = 1.0f / sqrtf(t);
    }
    __syncthreads();
    const float r = rinv_s;
    float2 o; o.x = v.x * r; o.y = v.y * r;
    ((float2*)(fn + (size_t)row * DD))[tid] = o;
}

// ---------------------------------------------------------------------------
// Kernel 2: per-sample focal + gender loss, and 7-bit validity mask.
// VALID table: g0==g1==0 -> {1,4} = 0x12 ; g0==g1==1 -> {2,5} = 0x24 ;
//              g0!=g1    -> {0,3,6} = 0x49
// ---------------------------------------------------------------------------
__global__ __launch_bounds__(256) void per_sample_kernel(
        const float* __restrict__ logits, const int* __restrict__ labels,
        const int* __restrict__ gf, const float* __restrict__ cw,
        float* __restrict__ partA, int* __restrict__ vbits) {
    const int i = blockIdx.x * blockDim.x + threadIdx.x;
    if (i >= BN) return;

    float l[NCLS];
    #pragma unroll
    for (int c = 0; c < NCLS; ++c) l[c] = logits[i * NCLS + c];

    float m = l[0];
    #pragma unroll
    for (int c = 1; c < NCLS; ++c) m = fmaxf(m, l[c]);
    float s = 0.0f;
    #pragma unroll
    for (int c = 0; c < NCLS; ++c) s += expf(l[c] - m);
    const float lse = m + logf(s);

    const int lab = labels[i];
    const float lp = l[lab] - lse;        // log p(label)
    const float ce = -lp;
    const float p  = expf(lp);
    const float om = 1.0f - p;
    const float weighted = cw[lab] * (om * om * ce);   // GAMMA = 2

    const int g0 = gf[2 * i], g1 = gf[2 * i + 1];
    const int vb = (g0 == g1) ? (g0 == 0 ? 0x12 : 0x24) : 0x49;

    float invalid_sum = 0.0f;
    #pragma unroll
    for (int c = 0; c < NCLS; ++c)
        if (!((vb >> c) & 1)) invalid_sum += expf(l[c] - lse);

    partA[i] = weighted + 2.0f * invalid_sum;
    vbits[i] = vb;
}

// ---------------------------------------------------------------------------
// Kernel 3: fused sim = fn @ fn^T (WMMA f32 16x16x4) + contrastive row sums.
// grid = 32 blocks x 256 threads (8 waves).  Wave w of block b owns row tile
// itile = b*8 + w (16 rows).  Each LDS stage holds TWO 16-col tiles (32x516
// floats = 66 KB); every A fragment load feeds two WMMAs -> halved A traffic
// and halved barrier count.  Epilogue is fully branchless (cndmask/fma only)
// so EXEC stays all-ones and nothing competes with the WMMA pipe via exec
// branching.
// Per-lane fragment pattern (identical for A and B per ISA layout):
//   lane L holds float2 at fn[tileBase + (L&15)][k + 2*(L>>4)]
// ---------------------------------------------------------------------------
__global__ __launch_bounds__(256) void sim_kernel(
        const float* __restrict__ fn, const int* __restrict__ labels,
        const int* __restrict__ vbits, float* __restrict__ mlpp) {
    __shared__ float ldsB[32 * LDS_STRIDE];   // two column tiles

    const int tid  = threadIdx.x;
    const int wave = tid >> 5;
    const int lane = tid & 31;
    const int lr   = lane & 15;   // column index within tile (for B/C)
    const int hi   = lane >> 4;   // half select
    const int itile   = blockIdx.x * 8 + wave;
    const int rowBase = itile * 16 + 8 * hi;  // first of this lane's 8 C rows

    int rowLab[8], rowVb[8];
    #pragma unroll
    for (int v = 0; v < 8; ++v) {
        rowLab[v] = labels[rowBase + v];
        rowVb[v]  = vbits[rowBase + v];
    }

    const float* aPtr = fn + (size_t)(itile * 16 + lr) * DD + 2 * hi;

    float accExp[8] = {}, accS[8] = {}, accC[8] = {};

    for (int jt = 0; jt < BN / 32; ++jt) {          // 128 double-tile stages
        __syncthreads();
        // cooperative stage of fn[jt*32 .. +31][0..511] into LDS
        {
            const float4* src = (const float4*)(fn + (size_t)jt * 32 * DD);
            #pragma unroll 4
            for (int idx = tid; idx < 32 * DD / 4; idx += 256) {
                const int row  = idx >> 7;       // (idx*4)/512
                const int col4 = idx & 127;
                float4 v = src[(size_t)row * (DD / 4) + col4];
                float* dst = &ldsB[row * LDS_STRIDE + col4 * 4];
                dst[0] = v.x; dst[1] = v.y; dst[2] = v.z; dst[3] = v.w;
            }
        }
        __syncthreads();

        const int jb0 = jt * 32;          // first subtile column base
        const int jb1 = jt * 32 + 16;     // second subtile column base
        const int colLab0 = labels[jb0 + lr];
        const int colLab1 = labels[jb1 + lr];

        v8f c0 = {}, c1 = {};
        const float* bPtr0 = &ldsB[lr * LDS_STRIDE + 2 * hi];
        const float* bPtr1 = bPtr0 + 16 * LDS_STRIDE;
        #pragma unroll 4
        for (int k = 0; k < DD; k += 4) {
            v2f a  = *(const v2f*)(aPtr  + k);
            v2f b0 = *(const v2f*)(bPtr0 + k);
            v2f b1 = *(const v2f*)(bPtr1 + k);
            c0 = __builtin_amdgcn_wmma_f32_16x16x4_f32(
                     false, a, false, b0, (short)0, c0, false, false);
            c1 = __builtin_amdgcn_wmma_f32_16x16x4_f32(
                     false, a, false, b1, (short)0, c1, false, false);
        }

        // branchless epilogue for both subtiles
        #pragma unroll
        for (int v = 0; v < 8; ++v) {
            const int r = rowBase + v;

            const float e0  = c0[v] * INV_T;
            const float ex0 = expf(e0);
            const float nd0 = (r != jb0 + lr) ? 1.0f : 0.0f;
            accExp[v] = fmaf(nd0, ex0, accExp[v]);
            const float mk0 = ((rowLab[v] == colLab0) &&
                               ((rowVb[v] >> colLab0) & 1)) ? 1.0f : 0.0f;
            accS[v] = fmaf(mk0, e0, accS[v]);
            accC[v] += mk0;

            const float e1  = c1[v] * INV_T;
            const float ex1 = expf(e1);
            const float nd1 = (r != jb1 + lr) ? 1.0f : 0.0f;
            accExp[v] = fmaf(nd1, ex1, accExp[v]);
            const float mk1 = ((rowLab[v] == colLab1) &&
                               ((rowVb[v] >> colLab1) & 1)) ? 1.0f : 0.0f;
            accS[v] = fmaf(mk1, e1, accS[v]);
            accC[v] += mk1;
        }
    }

    // reduce across the 16 column-lanes inside each half-wave
    #pragma unroll
    for (int v = 0; v < 8; ++v) {
        #pragma unroll
        for (int off = 1; off < 16; off <<= 1) {
            accExp[v] += __shfl_xor(accExp[v], off, 32);
            accS[v]   += __shfl_xor(accS[v],   off, 32);
            accC[v]   += __shfl_xor(accC[v],   off, 32);
        }
        if (lr == 0) {
            const float denom = fmaxf(accC[v], 1.0f);
            mlpp[rowBase + v] = -(accS[v] - accC[v] * logf(accExp[v])) / denom;
        }
    }
}

// ---------------------------------------------------------------------------
// Kernel 4: final scalar = mean(partA) + 0.1 * mean(mlpp)
// ---------------------------------------------------------------------------
__global__ __launch_bounds__(256) void final_kernel(
        const float* __restrict__ partA, const float* __restrict__ mlpp,
        float* __restrict__ out) {
    __shared__ float sA[256], sM[256];
    const int tid = threadIdx.x;
    float a = 0.0f, mm = 0.0f;
    for (int i = tid; i < BN; i += 256) { a += partA[i]; mm += mlpp[i]; }
    sA[tid] = a; sM[tid] = mm;
    __syncthreads();
    for (int s = 128; s > 0; s >>= 1) {
        if (tid < s) { sA[tid] += sA[tid + s]; sM[tid] += sM[tid + s]; }
        __syncthreads();
    }
    if (tid == 0) out[0] = sA[0] / (float)BN + 0.1f * (sM[0] / (float)BN);
}

// ---------------------------------------------------------------------------
extern "C" void kernel_launch(void* const* d_in, const int* in_sizes, int n_in,
                              void* d_out, int out_size, void* d_ws, size_t ws_size,
                              hipStream_t stream) {
    const float* logits  = (const float*)d_in[0];  // (4096, 7)
    const int*   labels  = (const int*)  d_in[1];  // (4096,)
    const int*   gf      = (const int*)  d_in[2];  // (4096, 2)
    const float* feats   = (const float*)d_in[3];  // (4096, 512)
    const float* cw      = (const float*)d_in[4];  // (7,)
    float*       out     = (float*)d_out;

    // workspace layout
    float* fn    = (float*)d_ws;                   // 4096*512 floats (8 MB)
    float* mlpp  = fn + (size_t)BN * DD;           // 4096 floats
    float* partA = mlpp + BN;                      // 4096 floats
    int*   vbits = (int*)(partA + BN);             // 4096 ints

    norm_kernel<<<BN, 256, 0, stream>>>(feats, fn);
    per_sample_kernel<<<BN / 256, 256, 0, stream>>>(logits, labels, gf, cw,
                                                    partA, vbits);
    sim_kernel<<<BN / 128, 256, 0, stream>>>(fn, labels, vbits, mlpp);
    final_kernel<<<1, 256, 0, stream>>>(partA, mlpp, out);
}